// APPNP_82824149336539
// MI455X (gfx1250) — compile-verified
//
#include <hip/hip_runtime.h>
#include <hip/hip_bf16.h>
#include <stdint.h>

// APPNP propagation on MI455X (gfx1250).
// Memory/atomic-bound sparse workload: WMMA does not apply (random-index
// segment sums). CDNA5-specific paths used: TDM (tensor_load_to_lds) async
// staging of edge streams into LDS + s_wait_tensorcnt, wave32 block sizing,
// fp32 L2 atomics.

#define DIM      48
#define DIM4     12           // DIM / 4 float4 chunks per node row
#define KSTEPS   5
#define ALPHA    0.8f
#define BETA     0.2f         // 1 - ALPHA
#define BLOCK    256          // 8 waves (wave32)
#define ECHUNK   256          // edges staged into LDS per block via TDM

typedef __attribute__((ext_vector_type(4))) unsigned int v4u;
typedef __attribute__((ext_vector_type(4))) int          v4i;
typedef __attribute__((ext_vector_type(8))) int          v8i;

// ---------------------------------------------------------------------------
// TDM: 1-D dword copy (cnt dwords) global -> LDS, per CDNA5 ISA ch.8 D# layout
// ---------------------------------------------------------------------------
__device__ __forceinline__ void tdm_load_dwords(void* lds_dst, const void* gsrc,
                                                int cnt) {
    uint32_t lds_addr = (uint32_t)(uintptr_t)lds_dst;        // LDS byte offset
    unsigned long long ga = (unsigned long long)(uintptr_t)gsrc;

    v4u g0;
    g0.x = 1u;                                   // count=1 (valid), user mode
    g0.y = lds_addr;                             // lds_addr [63:32]
    g0.z = (uint32_t)ga;                         // global_addr[31:0]
    g0.w = ((uint32_t)(ga >> 32) & 0x01FFFFFFu)  // global_addr[56:32]
           | (2u << 30);                         // type=2 ("image")

    v8i g1;
    g1[0] = (2 << 16);                 // workgroup_mask=0, data_size=2 (4B)
    g1[1] = (cnt & 0xFFFF) << 16;      // tensor_dim0[15:0] at bits[63:48]
    g1[2] = ((cnt >> 16) & 0xFFFF)     // tensor_dim0[31:16]
            | (1 << 16);               // tensor_dim1 = 1
    g1[3] = (cnt & 0xFFFF) << 16;      // tile_dim0 = cnt (single line)
    g1[4] = 0;                         // tile_dim1 = 0, tile_dim2 = 0
    g1[5] = cnt;                       // tensor_dim0_stride[31:0]
    g1[6] = 0;                         // stride hi, tensor_dim1_stride lo
    g1[7] = 0;                         // tensor_dim1_stride hi

    v4i z4 = {0, 0, 0, 0};
#if defined(__clang_major__) && (__clang_major__ >= 23)
    v8i z8 = {0, 0, 0, 0, 0, 0, 0, 0};
    __builtin_amdgcn_tensor_load_to_lds(g0, g1, z4, z4, z8, 0);
#else
    __builtin_amdgcn_tensor_load_to_lds(g0, g1, z4, z4, 0);
#endif
}

// ---------------------------------------------------------------------------
// Degree / normalization kernels
// ---------------------------------------------------------------------------
__global__ void deg_init_kernel(float* __restrict__ deg, int n) {
    int i = blockIdx.x * blockDim.x + threadIdx.x;
    if (i < n) deg[i] = 1.0f;                       // self-loop contribution
}

__global__ void deg_count_kernel(const int* __restrict__ col,
                                 float* __restrict__ deg, int e) {
    int i = blockIdx.x * blockDim.x + threadIdx.x;
    if (i < e) atomicAdd(&deg[col[i]], 1.0f);
}

__global__ void deg_to_dinv_kernel(float* __restrict__ deg, int n) {
    int i = blockIdx.x * blockDim.x + threadIdx.x;
    if (i < n) deg[i] = __frsqrt_rn(deg[i]);        // deg >= 1 always
}

__global__ void edge_w_kernel(const int* __restrict__ row,
                              const int* __restrict__ col,
                              const float* __restrict__ dinv,
                              float* __restrict__ w, int e) {
    int i = blockIdx.x * blockDim.x + threadIdx.x;
    if (i < e) w[i] = dinv[row[i]] * dinv[col[i]];
}

// ---------------------------------------------------------------------------
// acc = ALPHA * x + BETA * dinv^2 * h   (residual + self-loop term), float4
// ---------------------------------------------------------------------------
__global__ void acc_init_kernel(const float4* __restrict__ x4,
                                const float4* __restrict__ h4,
                                const float* __restrict__ dinv,
                                float4* __restrict__ acc4, int n) {
    int i = blockIdx.x * blockDim.x + threadIdx.x;  // over n * DIM4 chunks
    if (i >= n * DIM4) return;
    int node = i / DIM4;
    float d = dinv[node];
    float s = BETA * d * d;
    float4 xv = x4[i];
    float4 hv = h4[i];
    float4 o;
    o.x = ALPHA * xv.x + s * hv.x;
    o.y = ALPHA * xv.y + s * hv.y;
    o.z = ALPHA * xv.z + s * hv.z;
    o.w = ALPHA * xv.w + s * hv.w;
    acc4[i] = o;
}

// ---------------------------------------------------------------------------
// Edge scatter: acc[col] += BETA * w_e * h[row]
// Edge streams staged to LDS by the Tensor Data Mover (wave 0 issues DMA).
// 12 consecutive lanes cover one edge's 48 floats -> coalesced 192B gather.
// ---------------------------------------------------------------------------
__global__ void edge_scatter_kernel(const int* __restrict__ row,
                                    const int* __restrict__ col,
                                    const float* __restrict__ w,
                                    const float* __restrict__ h,
                                    float* __restrict__ acc, int e) {
    __shared__ int   s_row[ECHUNK];
    __shared__ int   s_col[ECHUNK];
    __shared__ float s_w[ECHUNK];

    int base = blockIdx.x * ECHUNK;
    int cnt  = e - base;
    if (cnt > ECHUNK) cnt = ECHUNK;

    if (threadIdx.x < 32) {   // one wave drives the TDM (EXEC ignored by TDM)
        tdm_load_dwords(s_row, row + base, cnt);
        tdm_load_dwords(s_col, col + base, cnt);
        tdm_load_dwords(s_w,   w   + base, cnt);
        __builtin_amdgcn_s_wait_tensorcnt(0);
    }
    __syncthreads();

    int total = cnt * DIM4;
    for (int i = threadIdx.x; i < total; i += BLOCK) {
        int ei = i / DIM4;
        int c  = i - ei * DIM4;
        int r  = s_row[ei];
        int d  = s_col[ei];
        float wf = BETA * s_w[ei];
        float4 hv = ((const float4*)(h + (size_t)r * DIM))[c];
        float* dst = acc + (size_t)d * DIM + c * 4;
        atomicAdd(dst + 0, wf * hv.x);
        atomicAdd(dst + 1, wf * hv.y);
        atomicAdd(dst + 2, wf * hv.z);
        atomicAdd(dst + 3, wf * hv.w);
    }
}

// ---------------------------------------------------------------------------
extern "C" void kernel_launch(void* const* d_in, const int* in_sizes, int n_in,
                              void* d_out, int out_size, void* d_ws, size_t ws_size,
                              hipStream_t stream) {
    const float* x   = (const float*)d_in[0];
    const int*   ei  = (const int*)d_in[1];
    const int n = in_sizes[0] / DIM;       // 100000
    const int e = in_sizes[1] / 2;         // 1600000
    const int* row = ei;                   // edge_index[0] = sources
    const int* col = ei + e;               // edge_index[1] = targets

    // Workspace carve-out (floats), 256B-aligned sections.
    float* dinv = (float*)d_ws;
    float* w    = dinv + (((size_t)n + 63) & ~(size_t)63);
    float* hB   = w    + (((size_t)e + 63) & ~(size_t)63);
    float* hA   = (float*)d_out;           // final step lands here (K odd)

    int gN  = (n + BLOCK - 1) / BLOCK;
    int gE  = (e + BLOCK - 1) / BLOCK;
    int gF  = (n * DIM4 + BLOCK - 1) / BLOCK;  // node-feature float4 grid
    int gEC = (e + ECHUNK - 1) / ECHUNK;       // edge-chunk grid

    // Normalization: deg -> dinv -> per-edge weights
    deg_init_kernel   <<<gN, BLOCK, 0, stream>>>(dinv, n);
    deg_count_kernel  <<<gE, BLOCK, 0, stream>>>(col, dinv, e);
    deg_to_dinv_kernel<<<gN, BLOCK, 0, stream>>>(dinv, n);
    edge_w_kernel     <<<gE, BLOCK, 0, stream>>>(row, col, dinv, w, e);

    // K propagation steps, ping-pong hA(d_out)/hB; step 0 reads x directly.
    const float* hprev = x;
    for (int k = 0; k < KSTEPS; ++k) {
        float* acc = (k & 1) ? hB : hA;
        acc_init_kernel<<<gF, BLOCK, 0, stream>>>(
            (const float4*)x, (const float4*)hprev, dinv, (float4*)acc, n);
        edge_scatter_kernel<<<gEC, BLOCK, 0, stream>>>(
            row, col, w, hprev, acc, e);
        hprev = acc;
    }
    // k=4 wrote hA == d_out.
}